// StarterAttentionSubModule_32212254720702
// MI455X (gfx1250) — compile-verified
//
#include <hip/hip_runtime.h>

typedef float v2f __attribute__((ext_vector_type(2)));
typedef float v8f __attribute__((ext_vector_type(8)));

namespace {
constexpr int XDIM = 329;
constexpr int NQ   = 25;
constexpr int KV   = 9;
constexpr int TILE = 16;            // samples per workgroup (one wave32)
constexpr int XPAD = 380;           // padded x row: 3*12 + 10*20 + 10*12 + 2*12

constexpr int SX_ELEMS = TILE * XPAD;              // 6080 (reused as out stage: 3600)
constexpr int SP_ELEMS = 4 * TILE * NQ * KV;       // 14400 (q,k,v,r planes)
constexpr int SW_ELEMS = 48*12 + 48*20 + 48*12 + 48*12;  // 2688 padded weights
constexpr int SB_ELEMS = 4 * 48;                   // 192 padded biases
constexpr int SMEM_ELEMS = SX_ELEMS + SP_ELEMS + SW_ELEMS + SB_ELEMS + NQ + 2 * KV;

constexpr int WOFF_J = 0;
constexpr int WOFF_O = 48 * 12;                    // 576
constexpr int WOFF_G = WOFF_O + 48 * 20;           // 1536
constexpr int WOFF_B = WOFF_G + 48 * 12;           // 2112
}

// original x column -> padded column (each sub-vector padded to a multiple of 4)
__device__ __forceinline__ int padcol(int c) {
  if (c < 27)  { int j = c / 9;          return        j * 12 + (c - j * 9); }
  if (c < 197) { int cc = c - 27;  int j = cc / 17; return  36 + j * 20 + (cc - j * 17); }
  if (c < 307) { int cc = c - 197; int j = cc / 11; return 236 + j * 12 + (cc - j * 11); }
  {            int cc = c - 307; int j = cc / 11; return 356 + j * 12 + (cc - j * 11); }
}

template <int L, int LPAD, int WOFF>
__device__ __forceinline__ void stage_w(const float* __restrict__ W, float* swp, int lane) {
  for (int i = lane; i < 4 * KV * L; i += 32) {
    const int n = i / L, k = i - n * L;
    swp[WOFF + n * LPAD + k] = W[i];
  }
}

// One group's Q/K/V/Res projections for the 16-sample tile via V_WMMA_F32_16X16X4_F32:
// C[16 x 36] = Xpad[16 x LPAD] * Wpad^T[LPAD x 48]; padding zeros make all loads unconditional.
template <int L, int NVEC, int COLP0, int Q0, int WOFF, int GIDX>
__device__ __forceinline__ void proj_group(const float* sxp, const float* swp,
                                           const float* sbias, float* sp,
                                           int m16, int hi) {
  constexpr int KSTEPS = (L + 3) / 4;
  constexpr int LPAD = KSTEPS * 4;
  #pragma unroll
  for (int nt = 0; nt < 3; ++nt) {
    const int nn = nt * 16 + m16;                  // output column, lane = N
    v2f bf[KSTEPS];                                // B frag: k = 4*ks + t + 2*hi
    #pragma unroll
    for (int ks = 0; ks < KSTEPS; ++ks)
      bf[ks] = *reinterpret_cast<const v2f*>(swp + WOFF + nn * LPAD + ks * 4 + 2 * hi);
    const float bias = sbias[GIDX * 48 + nn];

    for (int j = 0; j < NVEC; ++j) {
      const int colp = COLP0 + j * LPAD;
      v8f acc = {0.f, 0.f, 0.f, 0.f, 0.f, 0.f, 0.f, 0.f};
      #pragma unroll
      for (int ks = 0; ks < KSTEPS; ++ks) {
        // A frag: lane = M (sample), k = 4*ks + t + 2*hi ; aligned ds_load_b64
        const v2f a = *reinterpret_cast<const v2f*>(sxp + m16 * XPAD + colp + ks * 4 + 2 * hi);
        acc = __builtin_amdgcn_wmma_f32_16x16x4_f32(
            /*neg_a=*/false, a, /*neg_b=*/false, bf[ks],
            /*c_mod=*/(short)0, acc, /*reuse_a=*/false, /*reuse_b=*/false);
      }
      if (nn < 4 * KV) {
        const int p = nn / KV, kvc = nn - (nn / KV) * KV;
        const int q = Q0 + j;
        #pragma unroll
        for (int r = 0; r < 8; ++r) {              // C layout: M = r + 8*hi, N = lane&15
          const int m = r + 8 * hi;
          sp[((p * TILE + m) * NQ + q) * KV + kvc] = acc[r] + bias;
        }
      }
    }
  }
}

__global__ __launch_bounds__(32) void attn_fused_kernel(
    const float* __restrict__ x, const int* __restrict__ mask,
    const float* __restrict__ ln_g, const float* __restrict__ ln_b,
    const float* __restrict__ Wj, const float* __restrict__ bj,
    const float* __restrict__ Wo, const float* __restrict__ bo,
    const float* __restrict__ Wg, const float* __restrict__ bg,
    const float* __restrict__ Wb, const float* __restrict__ bb,
    float* __restrict__ out, int btot) {
  __shared__ __align__(16) float smem[SMEM_ELEMS];
  float* sxp   = smem;                       // [16][380] padded x, later out stage [16][225]
  float* sp    = smem + SX_ELEMS;            // [4][16][25][9]  q,k,v,r
  float* swp   = sp + SP_ELEMS;              // padded weights (48 x LPAD per group)
  float* sbias = swp + SW_ELEMS;             // [4][48] padded biases
  float* smask = sbias + SB_ELEMS;           // [25]
  float* sgam  = smask + NQ;                 // [9]
  float* sbet  = sgam + KV;                  // [9]

  const int lane = threadIdx.x;
  const long long tile0 = (long long)blockIdx.x * TILE;

  // ---- stage 0a: zero the padded staging regions ----
  for (int i = lane; i < SX_ELEMS; i += 32) sxp[i] = 0.0f;
  for (int i = lane; i < SW_ELEMS + SB_ELEMS; i += 32) swp[i] = 0.0f;  // swp + sbias contiguous
  __syncthreads();

  // ---- stage 0b: coalesced global reads, repack into padded layouts ----
  for (int i = lane; i < TILE * XDIM; i += 32) {
    const int s = i / XDIM, c = i - s * XDIM;
    const float vx = ((tile0 + s) < btot) ? x[tile0 * XDIM + i] : 0.0f;
    sxp[s * XPAD + padcol(c)] = vx;
  }
  stage_w< 9, 12, WOFF_J>(Wj, swp, lane);
  stage_w<17, 20, WOFF_O>(Wo, swp, lane);
  stage_w<11, 12, WOFF_G>(Wg, swp, lane);
  stage_w<11, 12, WOFF_B>(Wb, swp, lane);
  for (int i = lane; i < 4 * KV; i += 32) {
    sbias[0 * 48 + i] = bj[i];
    sbias[1 * 48 + i] = bo[i];
    sbias[2 * 48 + i] = bg[i];
    sbias[3 * 48 + i] = bb[i];
  }
  if (lane < NQ) smask[lane] = -1.0e9f * (float)mask[lane];
  if (lane < KV) { sgam[lane] = ln_g[lane]; sbet[lane] = ln_b[lane]; }
  __syncthreads();

  // ---- stage 1: projections via f32 WMMA ----
  const int m16 = lane & 15;
  const int hi  = lane >> 4;
  proj_group< 9,  3,   0,  0, WOFF_J, 0>(sxp, swp, sbias, sp, m16, hi);  // joint
  proj_group<17, 10,  36,  3, WOFF_O, 1>(sxp, swp, sbias, sp, m16, hi);  // object
  proj_group<11, 10, 236, 13, WOFF_G, 2>(sxp, swp, sbias, sp, m16, hi);  // goal
  proj_group<11,  2, 356, 23, WOFF_B, 3>(sxp, swp, sbias, sp, m16, hi);  // block
  __syncthreads();

  // ---- stage 2: per-sample attention + layernorm (2 lanes per sample) ----
  const int s = lane >> 1;
  const int h = lane & 1;
  const float* Pq = sp;
  const float* Pk = sp + 1 * TILE * NQ * KV;
  const float* Pv = sp + 2 * TILE * NQ * KV;
  const float* Pr = sp + 3 * TILE * NQ * KV;
  float* ob = sxp;  // reuse padded-x region as output staging [16][225]

  for (int qi = h; qi < NQ; qi += 2) {
    float qv[KV];
    #pragma unroll
    for (int d = 0; d < KV; ++d) qv[d] = Pq[(s * NQ + qi) * KV + d];

    float sc[NQ];
    float mx = -3.0e38f;
    #pragma unroll
    for (int j = 0; j < NQ; ++j) {
      float t = 0.0f;
      #pragma unroll
      for (int d = 0; d < KV; ++d) t += qv[d] * Pk[(s * NQ + j) * KV + d];
      t = t * (1.0f / 3.0f) + smask[j];          // /sqrt(9) + mask[key]*(-1e9)
      sc[j] = t;
      mx = fmaxf(mx, t);
    }
    float denom = 0.0f;
    #pragma unroll
    for (int j = 0; j < NQ; ++j) {
      const float e = __expf(sc[j] - mx);
      sc[j] = e;
      denom += e;
    }
    const float inv = 1.0f / denom;

    float o[KV];
    float mu = 0.0f;
    #pragma unroll
    for (int d = 0; d < KV; ++d) {
      float t = 0.0f;
      #pragma unroll
      for (int j = 0; j < NQ; ++j) t += sc[j] * Pv[(s * NQ + j) * KV + d];
      t = t * inv + Pr[(s * NQ + qi) * KV + d];
      o[d] = t;
      mu += t;
    }
    mu *= (1.0f / KV);
    float var = 0.0f;
    #pragma unroll
    for (int d = 0; d < KV; ++d) { const float dd = o[d] - mu; var += dd * dd; }
    var *= (1.0f / KV);
    const float rstd = rsqrtf(var + 1e-5f);
    #pragma unroll
    for (int d = 0; d < KV; ++d)
      ob[(s * NQ + qi) * KV + d] = (o[d] - mu) * rstd * sgam[d] + sbet[d];
  }
  __syncthreads();

  // ---- stage 3: coalesced store of [16][25][9] ----
  for (int i = lane; i < TILE * NQ * KV; i += 32) {
    const long long ss = tile0 + i / (NQ * KV);
    if (ss < btot) out[tile0 * NQ * KV + i] = ob[i];
  }
}

extern "C" void kernel_launch(void* const* d_in, const int* in_sizes, int n_in,
                              void* d_out, int out_size, void* d_ws, size_t ws_size,
                              hipStream_t stream) {
  (void)d_ws; (void)ws_size; (void)n_in; (void)out_size;
  const float* x    = (const float*)d_in[0];
  const int*   mask = (const int*)d_in[1];
  const float* ln_g = (const float*)d_in[2];
  const float* ln_b = (const float*)d_in[3];
  const float* Wj   = (const float*)d_in[4];
  const float* bj   = (const float*)d_in[5];
  const float* Wo   = (const float*)d_in[6];
  const float* bo   = (const float*)d_in[7];
  const float* Wg   = (const float*)d_in[8];
  const float* bg   = (const float*)d_in[9];
  const float* Wb   = (const float*)d_in[10];
  const float* bb   = (const float*)d_in[11];
  float* out = (float*)d_out;

  const int btot = in_sizes[0] / XDIM;
  const int grid = (btot + TILE - 1) / TILE;
  attn_fused_kernel<<<grid, 32, 0, stream>>>(x, mask, ln_g, ln_b,
                                             Wj, bj, Wo, bo, Wg, bg, Wb, bb,
                                             out, btot);
}